// ChamferLoss_70153995813078
// MI455X (gfx1250) — compile-verified
//
#include <hip/hip_runtime.h>

typedef float v2f __attribute__((ext_vector_type(2)));
typedef float v8f __attribute__((ext_vector_type(8)));

// Order-preserving float<->uint so unsigned atomicMin == float min.
__device__ __forceinline__ unsigned encOrd(float f) {
    unsigned u = __float_as_uint(f);
    return (u & 0x80000000u) ? ~u : (u | 0x80000000u);
}
__device__ __forceinline__ float decOrd(unsigned u) {
    unsigned b = (u & 0x80000000u) ? (u ^ 0x80000000u) : ~u;
    return __uint_as_float(b);
}

// Pack each cloud into WMMA operand layout, lo/hi halves contiguous so the
// base pointer stays uniform (SGPR) and the half-select is an element offset:
//   pack[i]     = (x, y)      -> K=0,1 half (lanes 0-15)
//   pack[n + i] = (z, |p|^2)  -> K=2,3 half (lanes 16-31)
// With A = (-2x,-2y,-2z,1), WMMA D[m][n] = |y_j|^2 - 2 x_i . y_j  directly.
__global__ __launch_bounds__(256) void chamfer_pack(
    const float* __restrict__ pred, const float* __restrict__ gt,
    v2f* __restrict__ Ppack, v2f* __restrict__ Gpack,
    unsigned* __restrict__ minP, unsigned* __restrict__ minG, int n)
{
    int i = blockIdx.x * blockDim.x + threadIdx.x;
    if (i >= n) return;
    float px = pred[3*i+0], py = pred[3*i+1], pz = pred[3*i+2];
    v2f t;
    t.x = px; t.y = py;                      Ppack[i] = t;
    t.x = pz; t.y = px*px + py*py + pz*pz;   Ppack[n + i] = t;
    float gx = gt[3*i+0], gy = gt[3*i+1], gz = gt[3*i+2];
    t.x = gx; t.y = gy;                      Gpack[i] = t;
    t.x = gz; t.y = gx*gx + gy*gy + gz*gz;   Gpack[n + i] = t;
    minP[i] = 0xFFFFFFFFu;
    minG[i] = 0xFFFFFFFFu;
}

// One wave per (16-row tile, column slice, direction).
__global__ __launch_bounds__(32) void chamfer_tiles(
    const v2f* __restrict__ Ppack, const v2f* __restrict__ Gpack,
    unsigned* __restrict__ minP, unsigned* __restrict__ minG,
    int n, int ntiles, int tilesPerSlice)
{
    const bool d0 = (blockIdx.z == 0);
    const v2f* Apack  = d0 ? Ppack : Gpack;
    const v2f* Bpack  = d0 ? Gpack : Ppack;
    unsigned*  outMin = d0 ? minP : minG;

    const int  lane = threadIdx.x;          // 0..31
    const int  ln   = lane & 15;
    const bool hi   = lane >= 16;           // K=2,3 half per ISA 16x4 f32 layout
    const unsigned halfOff = hi ? (unsigned)n : 0u;
    const int  i0   = blockIdx.x * 16;

    // A operand: lanes 0-15 -> (-2x, -2y); lanes 16-31 -> (-2z, 1)
    v2f av = Apack[halfOff + i0 + ln];
    v2f a;
    a.x = -2.0f * av.x;
    a.y = hi ? 1.0f : (-2.0f * av.y);

    v8f acc;
    #pragma unroll
    for (int v = 0; v < 8; ++v) acc[v] = 3.4e38f;

    int jt0 = blockIdx.y * tilesPerSlice;
    int jt1 = jt0 + tilesPerSlice;
    if (jt1 > ntiles) jt1 = ntiles;

    int jt = jt0;
    unsigned idx = halfOff + (unsigned)(jt0 * 16) + ln;   // element index, SGPR base + scaled voffset
    for (; jt + 3 < jt1; jt += 4, idx += 64u) {
        v2f b0 = Bpack[idx];
        v2f b1 = Bpack[idx + 16u];
        v2f b2 = Bpack[idx + 32u];
        v2f b3 = Bpack[idx + 48u];
        v8f c = {};
        v8f e0 = __builtin_amdgcn_wmma_f32_16x16x4_f32(false, a, false, b0, (short)0, c, false, false);
        v8f e1 = __builtin_amdgcn_wmma_f32_16x16x4_f32(false, a, false, b1, (short)0, c, false, false);
        v8f e2 = __builtin_amdgcn_wmma_f32_16x16x4_f32(false, a, false, b2, (short)0, c, false, false);
        v8f e3 = __builtin_amdgcn_wmma_f32_16x16x4_f32(false, a, false, b3, (short)0, c, false, false);
        #pragma unroll
        for (int v = 0; v < 8; ++v) {
            acc[v] = fminf(acc[v], fminf(e0[v], e1[v]));
            acc[v] = fminf(acc[v], fminf(e2[v], e3[v]));
        }
    }
    for (; jt < jt1; ++jt, idx += 16u) {
        v2f b0 = Bpack[idx];
        v8f c = {};
        v8f e0 = __builtin_amdgcn_wmma_f32_16x16x4_f32(false, a, false, b0, (short)0, c, false, false);
        #pragma unroll
        for (int v = 0; v < 8; ++v)
            acc[v] = fminf(acc[v], e0[v]);
    }

    // Min across each 16-lane half: lanes 0-15 carry rows 0..7, lanes 16-31 rows 8..15.
    #pragma unroll
    for (int mask = 1; mask < 16; mask <<= 1) {
        #pragma unroll
        for (int v = 0; v < 8; ++v) {
            float o = __shfl_xor(acc[v], mask, 16);
            acc[v] = fminf(acc[v], o);
        }
    }

    // d2 = |x_row|^2 + min(|y|^2 - 2 x.y); |x_row|^2 lives in Apack[n + row].y
    if (lane < 8) {
        int row = i0 + lane;
        atomicMin(&outMin[row], encOrd(Apack[n + row].y + acc[lane]));
    } else if (lane >= 24) {
        int row = i0 + lane - 16;
        atomicMin(&outMin[row], encOrd(Apack[n + row].y + acc[lane - 24]));
    }
}

__global__ __launch_bounds__(256) void chamfer_finish(
    const unsigned* __restrict__ minP, const unsigned* __restrict__ minG,
    float* __restrict__ out, int n)
{
    __shared__ float s[256];
    int tid = threadIdx.x;
    float invn = 1.0f / (float)n;
    float acc = 0.0f;
    for (int i = tid; i < n; i += 256) {
        float dp = decOrd(minP[i]);
        float dg = decOrd(minG[i]);
        acc += (sqrtf(fmaxf(dp, 0.0f)) + sqrtf(fmaxf(dg, 0.0f))) * invn;
    }
    s[tid] = acc;
    __syncthreads();
    for (int off = 128; off > 0; off >>= 1) {
        if (tid < off) s[tid] += s[tid + off];
        __syncthreads();
    }
    if (tid == 0) out[0] = s[0];
}

extern "C" void kernel_launch(void* const* d_in, const int* in_sizes, int n_in,
                              void* d_out, int out_size, void* d_ws, size_t ws_size,
                              hipStream_t stream) {
    const float* pred = (const float*)d_in[0];
    const float* gt   = (const float*)d_in[1];
    const int n = in_sizes[0] / 3;          // 16384 points per cloud

    char* ws = (char*)d_ws;
    v2f*      Ppack = (v2f*)(ws);                      // 2n v2f = n*16 bytes
    v2f*      Gpack = (v2f*)(ws + (size_t)n * 16);
    unsigned* minP  = (unsigned*)(ws + (size_t)n * 32);
    unsigned* minG  = (unsigned*)(ws + (size_t)n * 36);
    float* out = (float*)d_out;

    chamfer_pack<<<(n + 255) / 256, 256, 0, stream>>>(
        pred, gt, Ppack, Gpack, minP, minG, n);

    const int ntiles = n / 16;              // 1024 tiles of 16 points
    const int slices = 8;                   // 8 column slices -> 16384 waves total
    const int tps = (ntiles + slices - 1) / slices;
    dim3 grid(ntiles, slices, 2);           // z = direction (pred->gt, gt->pred)
    chamfer_tiles<<<grid, 32, 0, stream>>>(
        Ppack, Gpack, minP, minG, n, ntiles, tps);

    chamfer_finish<<<1, 256, 0, stream>>>(minP, minG, out, n);
}